// PairConvolution2_21191368639119
// MI455X (gfx1250) — compile-verified
//
#include <hip/hip_runtime.h>
#include <hip/hip_bf16.h>

typedef float v2f __attribute__((ext_vector_type(2)));
typedef float v8f __attribute__((ext_vector_type(8)));

static constexpr int B  = 4;
static constexpr int N  = 128;
static constexpr int CI = 32;
static constexpr int CO = 32;

// ---------------------------------------------------------------------------
// Stage 1: per (z,c) block.
//   FS[k]   = sum_d F[z,c,d,k]
//   FG[x,k] = sum_d geo[z,d,x] * F[z,c,d,k]
//   P[x,j]  = sum_k FG[x,k]*W1[x,j,k],  Q[x,j] = sum_k FS[k]*W1[x,j,k]
//   M1[z,c][0][j]   =  inv_norm * sum_x P[x,j]
//   M1[z,c][1+x][j] = -inv_norm * Q[x,j]
// 128 threads: lane k = t&31, d-group = t>>5 (4 groups of 32 d).
// ---------------------------------------------------------------------------
__global__ void stage1_build_M1(const float* __restrict__ F,
                                const float* __restrict__ G,
                                const float* __restrict__ W1,
                                const int*   __restrict__ nnorm,
                                float* __restrict__ M1) {
    const int z = blockIdx.x / N;
    const int c = blockIdx.x % N;
    const int t = threadIdx.x;
    const int k  = t & 31;
    const int dg = t >> 5;

    __shared__ float red[4][4][32];   // [dgroup][comp][k]
    __shared__ float sv[4][32];       // comp: 0=FS, 1..3=FG[x]
    __shared__ float sP[3][32];
    __shared__ float sQ[3][32];

    const float* Fzc = F + ((size_t)(z * N + c) * N) * CI;
    float fs = 0.f, fg0 = 0.f, fg1 = 0.f, fg2 = 0.f;
    for (int d = dg * 32; d < dg * 32 + 32; ++d) {
        const float v  = Fzc[d * CI + k];
        const float g0 = G[(z * N + d) * 3 + 0];   // scalar (uniform per wave)
        const float g1 = G[(z * N + d) * 3 + 1];
        const float g2 = G[(z * N + d) * 3 + 2];
        fs  += v;
        fg0 += g0 * v;
        fg1 += g1 * v;
        fg2 += g2 * v;
    }
    red[dg][0][k] = fs;
    red[dg][1][k] = fg0;
    red[dg][2][k] = fg1;
    red[dg][3][k] = fg2;
    __syncthreads();

    {   // reduce the 4 d-groups: thread t -> comp = t>>5, kk = t&31
        const int comp = t >> 5;
        const int kk   = t & 31;
        sv[comp][kk] = red[0][comp][kk] + red[1][comp][kk] +
                       red[2][comp][kk] + red[3][comp][kk];
    }
    __syncthreads();

    if (t < 96) {   // x = t/32, j = t%32
        const int x = t >> 5;
        const int j = t & 31;
        const float* w = W1 + (x * CO + j) * CI;
        float p = 0.f, q = 0.f;
        for (int kk = 0; kk < CI; ++kk) {
            p += sv[1 + x][kk] * w[kk];
            q += sv[0][kk]     * w[kk];
        }
        sP[x][j] = p;
        sQ[x][j] = q;
    }
    __syncthreads();

    const float inv = 1.0f / (float)nnorm[0];
    float* base = M1 + (size_t)(z * N + c) * 128;   // [4][32]
    if (t < 32) {
        base[t] = inv * (sP[0][t] + sP[1][t] + sP[2][t]);
    } else {
        const int x = (t >> 5) - 1;
        const int j = t & 31;
        base[(1 + x) * 32 + j] = -inv * sQ[x][j];
    }
}

// ---------------------------------------------------------------------------
// Stage 2: per (z,c) block, 8 waves. Wave w owns b-tile [16w, 16w+16).
//   tmp[z,c][b][j] = [1 | geo[z,b,:]]_{16x4}  x  M1[z,c]_{4x32}
// via V_WMMA_F32_16X16X4_F32 (one instruction per 16x16 output tile, K=4).
// ---------------------------------------------------------------------------
__global__ void stage2_wmma_tmp(const float* __restrict__ G,
                                const float* __restrict__ M1,
                                float* __restrict__ tmpbuf) {
    const int z    = blockIdx.x / N;
    const int c    = blockIdx.x % N;
    const int lane = threadIdx.x & 31;
    const int wave = threadIdx.x >> 5;

    // A fragment (16x4 f32): lanes 0-15 hold K0,K1 of row m; lanes 16-31 K2,K3.
    const int  m    = lane & 15;
    const bool loHf = lane < 16;
    const int  brow = wave * 16 + m;
    const float g0 = G[(z * N + brow) * 3 + 0];
    const float g1 = G[(z * N + brow) * 3 + 1];
    const float g2 = G[(z * N + brow) * 3 + 2];
    v2f a;
    a.x = loHf ? 1.0f : g1;
    a.y = loHf ? g0   : g2;

    const float* m1 = M1 + (size_t)(z * N + c) * 128;   // [4][32]
    const int n  = lane & 15;
    const int kb = loHf ? 0 : 2;   // B fragment: rows K0/K1 in lanes 0-15, K2/K3 in 16-31

    float* tb = tmpbuf + ((size_t)(z * N + c) * N) * CO;   // [b][j]
    const int rbase = wave * 16 + (loHf ? 0 : 8);

    #pragma unroll
    for (int jt = 0; jt < 2; ++jt) {
        v2f bf;
        bf.x = m1[kb * 32 + jt * 16 + n];
        bf.y = m1[(kb + 1) * 32 + jt * 16 + n];
        v8f cacc = {};
        v8f d = __builtin_amdgcn_wmma_f32_16x16x4_f32(
            false, a, false, bf, (short)0, cacc, false, false);
        #pragma unroll
        for (int v = 0; v < 8; ++v)
            tb[(rbase + v) * CO + jt * 16 + n] = d[v];
    }
}

// ---------------------------------------------------------------------------
// Stage 3: per (z,b) block.
//   TS[j]   = sum_c tmp[z,c,b,j]
//   TG[x,j] = sum_c geo[z,c,x] * tmp[z,c,b,j]
//   R[x,i]  = sum_j TG[x,j]*W2[x,i,j],  S[x,i] = sum_j TS[j]*W2[x,i,j]
//   M2[z,b][0][i] = sum_x R[x,i],  M2[z,b][1+x][i] = -S[x,i]
// ---------------------------------------------------------------------------
__global__ void stage3_build_M2(const float* __restrict__ tmpbuf,
                                const float* __restrict__ G,
                                const float* __restrict__ W2,
                                float* __restrict__ M2) {
    const int z = blockIdx.x / N;
    const int b = blockIdx.x % N;
    const int t = threadIdx.x;
    const int j  = t & 31;
    const int cg = t >> 5;

    __shared__ float red[4][4][32];
    __shared__ float sv[4][32];       // 0=TS, 1..3=TG[x]
    __shared__ float sR[3][32];
    __shared__ float sS[3][32];

    float ts = 0.f, tg0 = 0.f, tg1 = 0.f, tg2 = 0.f;
    for (int c = cg * 32; c < cg * 32 + 32; ++c) {
        const float v  = tmpbuf[((size_t)(z * N + c) * N + b) * CO + j];
        const float g0 = G[(z * N + c) * 3 + 0];
        const float g1 = G[(z * N + c) * 3 + 1];
        const float g2 = G[(z * N + c) * 3 + 2];
        ts  += v;
        tg0 += g0 * v;
        tg1 += g1 * v;
        tg2 += g2 * v;
    }
    red[cg][0][j] = ts;
    red[cg][1][j] = tg0;
    red[cg][2][j] = tg1;
    red[cg][3][j] = tg2;
    __syncthreads();

    {
        const int comp = t >> 5;
        const int jj   = t & 31;
        sv[comp][jj] = red[0][comp][jj] + red[1][comp][jj] +
                       red[2][comp][jj] + red[3][comp][jj];
    }
    __syncthreads();

    if (t < 96) {   // x = t/32, i = t%32
        const int x = t >> 5;
        const int i = t & 31;
        const float* w = W2 + (x * CO + i) * CO;
        float r = 0.f, s = 0.f;
        for (int jj = 0; jj < CO; ++jj) {
            r += sv[1 + x][jj] * w[jj];
            s += sv[0][jj]     * w[jj];
        }
        sR[x][i] = r;
        sS[x][i] = s;
    }
    __syncthreads();

    float* base = M2 + (size_t)(z * N + b) * 128;   // [4][32]
    if (t < 32) {
        base[t] = sR[0][t] + sR[1][t] + sR[2][t];
    } else {
        const int x = (t >> 5) - 1;
        const int i = t & 31;
        base[(1 + x) * 32 + i] = -sS[x][i];
    }
}

// ---------------------------------------------------------------------------
// Stage 4: per (z,b) block, 8 waves. Wave w owns a-tile [16w, 16w+16).
//   out[z][a][b][i] = [1 | geo[z,a,:]]_{16x4}  x  M2[z,b]_{4x32}
// ---------------------------------------------------------------------------
__global__ void stage4_wmma_out(const float* __restrict__ G,
                                const float* __restrict__ M2,
                                float* __restrict__ out) {
    const int z    = blockIdx.x / N;
    const int b    = blockIdx.x % N;
    const int lane = threadIdx.x & 31;
    const int wave = threadIdx.x >> 5;

    const int  m    = lane & 15;
    const bool loHf = lane < 16;
    const int  arow = wave * 16 + m;
    const float g0 = G[(z * N + arow) * 3 + 0];
    const float g1 = G[(z * N + arow) * 3 + 1];
    const float g2 = G[(z * N + arow) * 3 + 2];
    v2f a;
    a.x = loHf ? 1.0f : g1;
    a.y = loHf ? g0   : g2;

    const float* m2 = M2 + (size_t)(z * N + b) * 128;   // [4][32]
    const int n  = lane & 15;
    const int kb = loHf ? 0 : 2;
    const int rbase = wave * 16 + (loHf ? 0 : 8);

    #pragma unroll
    for (int jt = 0; jt < 2; ++jt) {
        v2f bf;
        bf.x = m2[kb * 32 + jt * 16 + n];
        bf.y = m2[(kb + 1) * 32 + jt * 16 + n];
        v8f cacc = {};
        v8f d = __builtin_amdgcn_wmma_f32_16x16x4_f32(
            false, a, false, bf, (short)0, cacc, false, false);
        #pragma unroll
        for (int v = 0; v < 8; ++v) {
            const int row = rbase + v;                       // a index
            out[((size_t)(z * N + row) * N + b) * CO + jt * 16 + n] = d[v];
        }
    }
}

extern "C" void kernel_launch(void* const* d_in, const int* in_sizes, int n_in,
                              void* d_out, int out_size, void* d_ws, size_t ws_size,
                              hipStream_t stream) {
    const float* F  = (const float*)d_in[0];   // [4,128,128,32]
    const float* G  = (const float*)d_in[1];   // [4,128,3]
    const float* W1 = (const float*)d_in[2];   // [3,32,32]
    const float* W2 = (const float*)d_in[3];   // [3,32,32]
    const int*   nn = (const int*)d_in[4];     // scalar n_norm

    float* out    = (float*)d_out;             // [4,128,128,32]
    float* ws     = (float*)d_ws;
    float* tmpbuf = ws;                        // 4*128*128*32 = 2,097,152 f
    float* M1     = tmpbuf + (size_t)B * N * N * CO;   // 4*128*4*32 = 65,536 f
    float* M2     = M1 + (size_t)B * N * 4 * CO;       // 65,536 f

    stage1_build_M1<<<B * N, 128, 0, stream>>>(F, G, W1, nn, M1);
    stage2_wmma_tmp<<<B * N, 256, 0, stream>>>(G, M1, tmpbuf);
    stage3_build_M2<<<B * N, 128, 0, stream>>>(tmpbuf, G, W2, M2);
    stage4_wmma_out<<<B * N, 256, 0, stream>>>(G, M2, out);
}